// FCOSDetect_90529320665597
// MI455X (gfx1250) — compile-verified
//
#include <hip/hip_runtime.h>
#include <math.h>

// ---------------- problem constants (match reference) ----------------
#define NB 64
#define NCLS 20
#define A_TOTAL 21824          // 128^2+64^2+32^2+16^2+8^2
#define NGRP 5456              // A_TOTAL / 4 (float4 groups per image)
#define MAXN 150
#define SCORE_THR_F 0.05f
#define IOU_THR_F 0.6f
#define TK_THREADS 1024
#define CHUNKS 6               // 6 chunks of 4096 anchors (last partial)
#define KSLOTS 24              // CHUNKS * 4 anchors per thread

#define GLOBAL_AS __attribute__((address_space(1)))
#define LDS_AS    __attribute__((address_space(3)))

struct Feats {
    const float* cls[5];
    const float* reg[5];
    const float* ctr[5];
};

__device__ __constant__ int d_hw[5]     = {128, 64, 32, 16, 8};
__device__ __constant__ int d_hwsq[5]   = {16384, 4096, 1024, 256, 64};
__device__ __constant__ int d_stride[5] = {8, 16, 32, 64, 128};

__device__ __forceinline__ int find_level(int idx, int* pos) {
    if (idx < 16384) { *pos = idx;          return 0; }
    if (idx < 20480) { *pos = idx - 16384;  return 1; }
    if (idx < 21504) { *pos = idx - 20480;  return 2; }
    if (idx < 21760) { *pos = idx - 21504;  return 3; }
    *pos = idx - 21760; return 4;
}

__device__ __forceinline__ float sigmoid_f(float x) {
    return 1.0f / (1.0f + expf(-x));
}

__device__ __forceinline__ unsigned long long pack_key(float sc, int idx) {
    // max key == highest score; ties -> lowest index (JAX top_k behavior).
    // Scores >= 0 so IEEE bits are monotone as unsigned.
    return ((unsigned long long)__float_as_uint(sc) << 32)
         | (unsigned long long)(0xFFFFFFFFu - (unsigned)idx);
}

// Compute the 4 scores for anchors [idx0, idx0+3] (idx0 % 4 == 0; a float4
// never crosses an FPN level boundary since all boundaries are multiples of 4).
__device__ __forceinline__ float4 score4(const Feats& F, int b, int idx0) {
    int pos0;
    const int li   = find_level(idx0, &pos0);
    const int hwsq = d_hwsq[li];
    const int hq4  = hwsq >> 2;                // channel stride in float4s

    const float4* cp = (const float4*)(F.cls[li] + (size_t)b * NCLS * hwsq) + (pos0 >> 2);
    float4 m = cp[0];
    #pragma unroll
    for (int c = 1; c < NCLS; ++c) {
        const float4 v = cp[(size_t)c * hq4];
        m.x = fmaxf(m.x, v.x);
        m.y = fmaxf(m.y, v.y);
        m.z = fmaxf(m.z, v.z);
        m.w = fmaxf(m.w, v.w);
    }
    const float4 ct = *((const float4*)(F.ctr[li] + (size_t)b * hwsq) + (pos0 >> 2));
    float4 r;
    r.x = sqrtf(sigmoid_f(m.x) * sigmoid_f(ct.x));
    r.y = sqrtf(sigmoid_f(m.y) * sigmoid_f(ct.y));
    r.z = sqrtf(sigmoid_f(m.z) * sigmoid_f(ct.z));
    r.w = sqrtf(sigmoid_f(m.w) * sigmoid_f(ct.w));
    return r;
}

// ==========================================================================
// Shared top-150 selection over KSLOTS register keys (one 1024-thread block
// per image). Key mapping: idx = chunk*4096 + 4*t + e, slot = chunk*4 + e.
// The per-thread local max is cached in a register and only recomputed by the
// single thread whose slot was consumed -> ~30 VALU/round instead of ~100.
// ==========================================================================
__device__ __forceinline__ void topk_select(unsigned long long (&key)[KSLOTS],
                                            unsigned long long* red,
                                            int b, int t, int lane, int wid,
                                            float* __restrict__ tsc,
                                            int* __restrict__ tix) {
    unsigned long long mylocal = 0ULL;
    #pragma unroll
    for (int k = 0; k < KSLOTS; ++k) mylocal = (key[k] > mylocal) ? key[k] : mylocal;

    for (int it = 0; it < MAXN; ++it) {
        // wave32 reduction of cached local maxima
        unsigned long long best = mylocal;
        #pragma unroll
        for (int off = 16; off > 0; off >>= 1) {
            unsigned long long o = __shfl_down(best, off, 32);
            best = (o > best) ? o : best;
        }
        if (lane == 0) red[wid] = best;
        __syncthreads();
        if (wid == 0) {
            unsigned long long v = red[lane];   // exactly 32 waves
            #pragma unroll
            for (int off = 16; off > 0; off >>= 1) {
                unsigned long long o = __shfl_down(v, off, 32);
                v = (o > v) ? o : v;
            }
            if (lane == 0) red[0] = v;
        }
        __syncthreads();
        const unsigned long long win = red[0];
        const unsigned widx = 0xFFFFFFFFu - (unsigned)(win & 0xFFFFFFFFull);
        if (t == 0) {
            tsc[b * MAXN + it] = __uint_as_float((unsigned)(win >> 32));
            tix[b * MAXN + it] = (int)widx;
        }
        // owning thread clears the selected slot and refreshes its local max
        if ((int)((widx & 4095u) >> 2) == t) {
            const int ks = (int)(((widx >> 12) << 2) | (widx & 3u));
            #pragma unroll
            for (int k = 0; k < KSLOTS; ++k) if (k == ks) key[k] = 0ULL;
            mylocal = 0ULL;
            #pragma unroll
            for (int k = 0; k < KSLOTS; ++k) mylocal = (key[k] > mylocal) ? key[k] : mylocal;
        }
        __syncthreads();
    }
}

// ==========================================================================
// Split path, kernel 1a: grid-wide scoring (full chip occupancy; this kernel
// moves ~95% of all bytes). One aligned float4 of scores per thread, written
// contiguously per image: scores[b*A_TOTAL + idx].
// ==========================================================================
__global__ __launch_bounds__(256)
void fcos_score_kernel(Feats F, float* __restrict__ scores) {
    const int g = blockIdx.x * 256 + threadIdx.x;   // float4 group id, < NB*NGRP
    const int b = g / NGRP;
    const int q = g - b * NGRP;
    ((float4*)scores)[g] = score4(F, b, q << 2);
}

// ==========================================================================
// Split path, kernel 1b: per-image exact top-150 from the contiguous score
// array (87 KB/image, L2-resident given 192 MB L2).
// ==========================================================================
__global__ __launch_bounds__(TK_THREADS)
void fcos_topk_kernel(const float* __restrict__ scores,
                      float* __restrict__ tsc, int* __restrict__ tix) {
    const int b    = blockIdx.x;
    const int t    = threadIdx.x;
    const int lane = t & 31;
    const int wid  = t >> 5;

    __shared__ unsigned long long red[TK_THREADS / 32];

    unsigned long long key[KSLOTS];
    #pragma unroll
    for (int s = 0; s < CHUNKS; ++s) {
        #pragma unroll
        for (int e = 0; e < 4; ++e) key[s * 4 + e] = 0ULL;
        const int q = s * 1024 + t;                 // float4 group within image
        if (q < NGRP) {
            const float4 v = ((const float4*)scores)[b * NGRP + q];
            const int idx0 = q << 2;                // == s*4096 + 4*t
            key[s * 4 + 0] = pack_key(v.x, idx0 + 0);
            key[s * 4 + 1] = pack_key(v.y, idx0 + 1);
            key[s * 4 + 2] = pack_key(v.z, idx0 + 2);
            key[s * 4 + 3] = pack_key(v.w, idx0 + 3);
        }
    }
    topk_select(key, red, b, t, lane, wid, tsc, tix);
}

// ==========================================================================
// Fallback path (small workspace): fused scoring + top-150, one block/image.
// ==========================================================================
__global__ __launch_bounds__(TK_THREADS)
void fcos_score_topk_kernel(Feats F, float* __restrict__ tsc, int* __restrict__ tix) {
    const int b    = blockIdx.x;
    const int t    = threadIdx.x;
    const int lane = t & 31;
    const int wid  = t >> 5;

    __shared__ unsigned long long red[TK_THREADS / 32];

    unsigned long long key[KSLOTS];
    #pragma unroll
    for (int ch = 0; ch < CHUNKS; ++ch) {
        #pragma unroll
        for (int e = 0; e < 4; ++e) key[ch * 4 + e] = 0ULL;
        const int idx0 = ch * 4096 + (t << 2);
        if (idx0 < A_TOTAL) {
            const float4 r = score4(F, b, idx0);
            key[ch * 4 + 0] = pack_key(r.x, idx0 + 0);
            key[ch * 4 + 1] = pack_key(r.y, idx0 + 1);
            key[ch * 4 + 2] = pack_key(r.z, idx0 + 2);
            key[ch * 4 + 3] = pack_key(r.w, idx0 + 3);
        }
    }
    topk_select(key, red, b, t, lane, wid, tsc, tix);
}

// ==========================================================================
// Kernel 2: decode the 150 selected boxes, class id, greedy NMS, outputs.
// Top-150 table is staged into LDS via the gfx1250 async global->LDS path.
// ==========================================================================
__global__ __launch_bounds__(256)
void fcos_nms_kernel(Feats F, const float* __restrict__ tsc, const int* __restrict__ tix,
                     float* __restrict__ out) {
    const int b = blockIdx.x;
    const int j = threadIdx.x;

    __shared__ float s_ts[MAXN];
    __shared__ int   s_ti[MAXN];
    __shared__ float bx1[MAXN], by1[MAXN], bx2[MAXN], by2[MAXN];
    __shared__ float sarea[MAXN], soffs[MAXN], sscore[MAXN], scmax[MAXN];
    __shared__ int   sid_[MAXN], skeep[MAXN];
    __shared__ float s_cm;

    // ---- stage (score, idx) into LDS (async on gfx1250, fallback otherwise) ----
#if __has_builtin(__builtin_amdgcn_global_load_async_to_lds_b32)
    if (j < MAXN) {
        __builtin_amdgcn_global_load_async_to_lds_b32(
            (GLOBAL_AS int*)(int*)(void*)(tsc + b * MAXN + j),
            (LDS_AS int*)(int*)(void*)&s_ts[j], 0, 0);
        __builtin_amdgcn_global_load_async_to_lds_b32(
            (GLOBAL_AS int*)(int*)(void*)(tix + b * MAXN + j),
            (LDS_AS int*)(int*)(void*)&s_ti[j], 0, 0);
    }
#if __has_builtin(__builtin_amdgcn_s_wait_asynccnt)
    __builtin_amdgcn_s_wait_asynccnt(0);
#else
    asm volatile("s_wait_asynccnt 0" ::: "memory");
#endif
#else
    if (j < MAXN) {
        s_ts[j] = tsc[b * MAXN + j];
        s_ti[j] = tix[b * MAXN + j];
    }
#endif
    __syncthreads();

    // ---- decode box + class id for my candidate ----
    if (j < MAXN) {
        const float sc  = s_ts[j];
        const int   idx = s_ti[j];
        int pos;
        const int li   = find_level(idx, &pos);
        const int hwsq = d_hwsq[li];
        const int w    = d_hw[li];
        const int st   = d_stride[li];
        const int iy   = pos / w;
        const int ix   = pos - iy * w;
        const float cx = (float)(ix * st + (st >> 1));
        const float cy = (float)(iy * st + (st >> 1));
        const float* rp = F.reg[li] + (size_t)b * 4 * hwsq + pos;
        const float fs = (float)st;
        const float o0 = rp[0] * fs;
        const float o1 = rp[(size_t)hwsq] * fs;
        const float o2 = rp[(size_t)2 * hwsq] * fs;
        const float o3 = rp[(size_t)3 * hwsq] * fs;
        const float x1 = cx - o0, y1 = cy - o1, x2 = cx + o2, y2 = cy + o3;

        // argmax over raw logits == argmax over sigmoids (monotone), first wins on ties
        const float* cp = F.cls[li] + (size_t)b * NCLS * hwsq + pos;
        float m = cp[0];
        int am = 0;
        #pragma unroll
        for (int c = 1; c < NCLS; ++c) {
            const float v = cp[(size_t)c * hwsq];
            if (v > m) { m = v; am = c; }
        }
        const int id = am + 1;
        const int kp = (sc > SCORE_THR_F) ? 1 : 0;

        bx1[j] = x1; by1[j] = y1; bx2[j] = x2; by2[j] = y2;
        sscore[j] = sc; sid_[j] = id; skeep[j] = kp;
        sarea[j] = (x2 - x1 + 1.0f) * (y2 - y1 + 1.0f);     // offset cancels in area
        const float cmx = fmaxf(fmaxf(x1, y1), fmaxf(x2, y2));
        scmax[j] = kp ? cmx : -INFINITY;
    }
    __syncthreads();

    // coord_max over valid boxes (reference: -inf -> 0 when none valid)
    if (j == 0) {
        float cm = -INFINITY;
        for (int i = 0; i < MAXN; ++i) cm = fmaxf(cm, scmax[i]);
        if (cm == -INFINITY) cm = 0.0f;
        s_cm = cm;
    }
    __syncthreads();
    if (j < MAXN) soffs[j] = (float)sid_[j] * (s_cm + 1.0f);
    __syncthreads();

    // ---- greedy NMS, exact reference order: keep[i] read per round, suppress j>i ----
    for (int i = 0; i < MAXN; ++i) {
        if (skeep[i] && j > i && j < MAXN && skeep[j]) {
            const float oi = soffs[i], oj = soffs[j];
            const float xmin = fmaxf(bx1[i] + oi, bx1[j] + oj);
            const float ymin = fmaxf(by1[i] + oi, by1[j] + oj);
            const float xmax = fminf(bx2[i] + oi, bx2[j] + oj);
            const float ymax = fminf(by2[i] + oi, by2[j] + oj);
            const float ow = fmaxf(xmax - xmin, 0.0f);
            const float oh = fmaxf(ymax - ymin, 0.0f);
            const float overlap = ow * oh;
            const float uni = fmaxf(sarea[i] + sarea[j] - overlap, 1e-10f);
            if (overlap / uni > IOU_THR_F) skeep[j] = 0;
        }
        __syncthreads();
    }

    // ---- outputs: scores | ids | boxes | keep (flat float) ----
    if (j < MAXN) {
        const int kp = skeep[j];
        const int g  = b * MAXN + j;
        out[g] = kp ? sscore[j] : 0.0f;                       // out_scores
        out[NB * MAXN + g] = kp ? (float)sid_[j] : 0.0f;      // out_ids
        float* ob = out + 2 * NB * MAXN + (size_t)g * 4;      // out_boxes
        ob[0] = kp ? bx1[j] : 0.0f;
        ob[1] = kp ? by1[j] : 0.0f;
        ob[2] = kp ? bx2[j] : 0.0f;
        ob[3] = kp ? by2[j] : 0.0f;
        out[2 * NB * MAXN + NB * MAXN * 4 + g] = kp ? 1.0f : 0.0f;  // keep
    }
}

// ==========================================================================
extern "C" void kernel_launch(void* const* d_in, const int* in_sizes, int n_in,
                              void* d_out, int out_size, void* d_ws, size_t ws_size,
                              hipStream_t stream) {
    (void)in_sizes; (void)n_in; (void)out_size;

    // setup_inputs() dict order: cls0, reg0, ctr0, cls1, reg1, ctr1, ...
    Feats F;
    for (int li = 0; li < 5; ++li) {
        F.cls[li] = (const float*)d_in[3 * li + 0];
        F.reg[li] = (const float*)d_in[3 * li + 1];
        F.ctr[li] = (const float*)d_in[3 * li + 2];
    }

    // workspace layout: [ tsc: NB*MAXN f32 | tix: NB*MAXN i32 | scores: NB*A f32 ]
    const size_t top_bytes = (size_t)NB * MAXN * (sizeof(float) + sizeof(int));
    const size_t big_bytes = top_bytes + (size_t)NB * A_TOTAL * sizeof(float);

    float* tsc = (float*)d_ws;
    int*   tix = (int*)((char*)d_ws + (size_t)NB * MAXN * sizeof(float));

    if (ws_size >= big_bytes) {
        // split path: full-occupancy streaming, then L2-resident top-k
        float* scores = (float*)((char*)d_ws + top_bytes);   // 16B-aligned (76800 % 16 == 0)
        fcos_score_kernel<<<(NB * NGRP) / 256, 256, 0, stream>>>(F, scores);
        fcos_topk_kernel<<<NB, TK_THREADS, 0, stream>>>(scores, tsc, tix);
    } else {
        // fallback: fused scoring + top-k, one block per image
        fcos_score_topk_kernel<<<NB, TK_THREADS, 0, stream>>>(F, tsc, tix);
    }
    fcos_nms_kernel<<<NB, 256, 0, stream>>>(F, tsc, tix, (float*)d_out);
}